// speMultiAttn_9165460210059
// MI455X (gfx1250) — compile-verified
//
#include <hip/hip_runtime.h>

// speMultiAttn for MI455X (gfx1250, wave32).
// Shapes fixed by the reference: B=8, C=128, H=W=256, HEADS=8, P=8 (P2=64), K=3.
// Algebra: attn[b,c,s] = A[b,h(s)] @ x[b,:,s] + beta[b,h(s)],
//   A = diag(1/denom) * q * (k^T * Wv),  beta = diag(1/denom) * q * (k^T * bv),
//   denom = q @ ksum + eps  (the reference's else-branch of _linear_attn).
// The big pass (17.2 GFLOP GEMM over 268 MB of x) uses V_WMMA_F32_16X16X4_F32.

#define BN      8
#define CN      128
#define HN      256
#define WN      256
#define HWN     65536
#define HEADSN  8
#define P2N     64
#define D2N     8192      // HW / HEADS (= 32 image rows)
#define EPSF    1e-7f

typedef float v2f __attribute__((ext_vector_type(2)));
typedef float v8f __attribute__((ext_vector_type(8)));

// ---------------------------------------------------------------------------
// Stage 1: adaptive avg-pool 256x256 -> 8x8 (mean of 32x32 blocks).
// One block per (b,c); thread t owns image column t; wave w owns column-block w.
// ---------------------------------------------------------------------------
__global__ __launch_bounds__(256) void k_pool(const float* __restrict__ x,
                                              float* __restrict__ pooled) {
  int bc = blockIdx.x;                       // 0..B*C-1
  const float* img = x + (size_t)bc * HWN;
  int t = threadIdx.x;                       // column 0..255
  float acc[8] = {0.f,0.f,0.f,0.f,0.f,0.f,0.f,0.f};
  for (int y = 0; y < HN; ++y)               // coalesced full-row reads
    acc[y >> 5] += img[(size_t)y * WN + t];
  int lane = t & 31, px = t >> 5;            // wave == column block
#pragma unroll
  for (int py = 0; py < 8; ++py) {
    float v = acc[py];
    v += __shfl_down(v, 16, 32);
    v += __shfl_down(v,  8, 32);
    v += __shfl_down(v,  4, 32);
    v += __shfl_down(v,  2, 32);
    v += __shfl_down(v,  1, 32);
    if (lane == 0)
      pooled[(size_t)bc * 64 + py * 8 + px] = v * (1.0f / 1024.0f);
  }
}

// ---------------------------------------------------------------------------
// Stage 2: q = relu(pooled @ Wq^T + bq), k = relu(pooled @ Wk^T + bk)
// pooled: [1024,64]; Wq/Wk: [512,64]; out q/k: [1024,512]. Tiny; one thread/out.
// ---------------------------------------------------------------------------
__global__ __launch_bounds__(256) void k_qk(const float* __restrict__ pooled,
                                            const float* __restrict__ Wq,
                                            const float* __restrict__ bq,
                                            const float* __restrict__ Wk,
                                            const float* __restrict__ bk,
                                            float* __restrict__ qb,
                                            float* __restrict__ kb) {
  int idx = blockIdx.x * 256 + threadIdx.x;  // 0 .. 1024*512-1
  int row = idx >> 9;                        // bc
  int o   = idx & 511;                       // head*64 + p
  const float* pr = pooled + (size_t)row * 64;
  const float* wq = Wq + (size_t)o * 64;
  const float* wk = Wk + (size_t)o * 64;
  float sq = bq[o], sk = bk[o];
#pragma unroll 8
  for (int j = 0; j < 64; ++j) {
    float p = pr[j];
    sq += p * wq[j];
    sk += p * wk[j];
  }
  qb[idx] = fmaxf(sq, 0.0f);
  kb[idx] = fmaxf(sk, 0.0f);
}

// ---------------------------------------------------------------------------
// Stage 3: per (b,h): T = k^T Wv (64x128, LDS), denom, beta, A = diag(1/denom) q T.
// One 256-thread block per (b,h). ~34 KB static LDS.
// ---------------------------------------------------------------------------
__global__ __launch_bounds__(256) void k_prep(const float* __restrict__ qb,
                                              const float* __restrict__ kb,
                                              const float* __restrict__ Wv,
                                              const float* __restrict__ bv,
                                              float* __restrict__ Ab,
                                              float* __restrict__ betab) {
  __shared__ float T[64 * 128];
  __shared__ float ksum[64];
  __shared__ float tb[64];
  __shared__ float dn[128];
  int bh = blockIdx.x;
  int b = bh >> 3, h = bh & 7;
  int tid = threadIdx.x;
  const float* kbase = kb + (size_t)b * CN * 512 + h * 64;  // + c*512 + p
  const float* qbase = qb + (size_t)b * CN * 512 + h * 64;

  if (tid < 64) {                       // ksum[p], tb[p] = sum_c k[c,p]*bv[c]
    float s = 0.f, sb = 0.f;
    for (int c = 0; c < CN; ++c) {
      float kv = kbase[(size_t)c * 512 + tid];
      s += kv;
      sb += kv * bv[c];
    }
    ksum[tid] = s;
    tb[tid] = sb;
  }
  for (int i = tid; i < 64 * 128; i += 256) {   // T[p,cc] = sum_c k[c,p]*Wv[c,cc]
    int p = i >> 7, cc = i & 127;
    float s = 0.f;
    for (int c = 0; c < CN; ++c)
      s += kbase[(size_t)c * 512 + p] * Wv[(size_t)c * 128 + cc];
    T[i] = s;
  }
  __syncthreads();
  if (tid < 128) {                      // denom[c], beta[c]
    const float* qrow = qbase + (size_t)tid * 512;
    float s = EPSF, sb = 0.f;
    for (int p = 0; p < 64; ++p) {
      float qv = qrow[p];
      s += qv * ksum[p];
      sb += qv * tb[p];
    }
    dn[tid] = s;
    betab[bh * 128 + tid] = sb / s;
  }
  __syncthreads();
  for (int i = tid; i < 128 * 128; i += 256) {  // A[c,cc]
    int cr = i >> 7, cc = i & 127;
    const float* qrow = qbase + (size_t)cr * 512;
    float s = 0.f;
    for (int p = 0; p < 64; ++p)
      s += qrow[p] * T[p * 128 + cc];
    Ab[(size_t)bh * (128 * 128) + i] = s / dn[cr];
  }
}

// ---------------------------------------------------------------------------
// Stage 4: attn[b,:,chunk] = A[b,h] (128x128) @ X (128x8192) + beta, per (b,h).
// WMMA f32 16x16x4. Grid (128 col-tiles, 64 bh). Block = 256 = 8 waves.
// Each workgroup: 128x64 output tile; X panel (128Kx64N) staged in LDS.
// Wave w -> rows 16w..16w+15 with 4 accumulators (16x64).
// A-frag (16x4 f32): lane l holds A[m0+l%16][k0 + 2*(l/16) + j], j=0,1 (float2).
// B-frag (4x16 f32): lane l holds X[k0 + 2*(l/16) + j][n0 + l%16].
// ---------------------------------------------------------------------------
__global__ __launch_bounds__(256) void k_attn(const float* __restrict__ x,
                                              const float* __restrict__ Ab,
                                              const float* __restrict__ betab,
                                              float* __restrict__ attn) {
  __shared__ __align__(16) float Xs[128 * 68];   // padded stride 68
  int n0 = blockIdx.x * 64;
  int bh = blockIdx.y;
  int b = bh >> 3, h = bh & 7;
  const size_t xbase = (size_t)b * CN * HWN + (size_t)h * D2N + n0;

  // Stage X panel: 128 channels x 64 pixels, float4 coalesced (uniform loop).
  for (int i = threadIdx.x; i < 128 * 16; i += 256) {
    int c = i >> 4, j4 = i & 15;
    const float4 vv = *(const float4*)(x + xbase + (size_t)c * HWN + j4 * 4);
    *(float4*)(&Xs[c * 68 + j4 * 4]) = vv;
  }
  __syncthreads();

  int wave = threadIdx.x >> 5, lane = threadIdx.x & 31;
  int m0 = wave * 16;
  int lm = lane & 15, lh = lane >> 4;
  const float* Arow = Ab + (size_t)bh * (128 * 128) + (size_t)(m0 + lm) * 128 + 2 * lh;

  v8f acc[4] = {};
  for (int kk = 0; kk < 32; ++kk) {              // K = 128 in steps of 4
    v2f a = *(const v2f*)(Arow + 4 * kk);        // contiguous pair, 8B aligned
    int kr = 4 * kk + 2 * lh;
#pragma unroll
    for (int nt = 0; nt < 4; ++nt) {
      v2f bb;
      bb.x = Xs[kr * 68 + nt * 16 + lm];
      bb.y = Xs[(kr + 1) * 68 + nt * 16 + lm];
      acc[nt] = __builtin_amdgcn_wmma_f32_16x16x4_f32(
          /*neg_a=*/false, a, /*neg_b=*/false, bb,
          /*c_mod=*/(short)0, acc[nt], /*reuse_a=*/false, /*reuse_b=*/false);
    }
  }

  // Epilogue: +beta, store. D layout: vgpr v, lane l -> D[v + 8*(l/16)][l%16].
  size_t obase = (size_t)b * CN * HWN + (size_t)h * D2N + n0;
#pragma unroll
  for (int nt = 0; nt < 4; ++nt) {
#pragma unroll
    for (int v = 0; v < 8; ++v) {
      int r = m0 + v + 8 * lh;
      attn[obase + (size_t)r * HWN + nt * 16 + lm] = acc[nt][v] + betab[bh * 128 + r];
    }
  }
}

// ---------------------------------------------------------------------------
// Stage 5: depthwise 3x3 conv (pad 1) + bp. 4 pixels per thread, float4 store.
// ---------------------------------------------------------------------------
__global__ __launch_bounds__(256) void k_dw(const float* __restrict__ attn,
                                            const float* __restrict__ Wp,
                                            const float* __restrict__ bp,
                                            float* __restrict__ out) {
  size_t idx = (size_t)blockIdx.x * 256 + threadIdx.x;  // quad index
  int x0 = (int)(idx & 63) * 4;
  int y  = (int)((idx >> 6) & 255);
  int bc = (int)(idx >> 14);
  int c = bc & 127;
  const float* img = attn + (size_t)bc * HWN;
  const float* wp = Wp + c * 9;

  float row[3][6];
#pragma unroll
  for (int dy = 0; dy < 3; ++dy) {
    int yy = y + dy - 1;
    bool yok = (yy >= 0) && (yy < HN);
#pragma unroll
    for (int j = 0; j < 6; ++j) {
      int xc = x0 + j - 1;
      bool ok = yok && (xc >= 0) && (xc < WN);
      row[dy][j] = ok ? img[(size_t)yy * WN + xc] : 0.0f;
    }
  }
  float b0 = bp[c];
  float o[4];
#pragma unroll
  for (int i = 0; i < 4; ++i) {
    float s = b0;
#pragma unroll
    for (int dy = 0; dy < 3; ++dy)
#pragma unroll
      for (int dx = 0; dx < 3; ++dx)
        s += wp[dy * 3 + dx] * row[dy][i + dx];
    o[i] = s;
  }
  float4 r4 = make_float4(o[0], o[1], o[2], o[3]);
  *(float4*)(out + (size_t)bc * HWN + (size_t)y * WN + x0) = r4;
}

// ---------------------------------------------------------------------------
extern "C" void kernel_launch(void* const* d_in, const int* in_sizes, int n_in,
                              void* d_out, int out_size, void* d_ws, size_t ws_size,
                              hipStream_t stream) {
  (void)in_sizes; (void)n_in; (void)out_size; (void)ws_size;
  const float* x  = (const float*)d_in[0];
  const float* Wq = (const float*)d_in[1];
  const float* bq = (const float*)d_in[2];
  const float* Wk = (const float*)d_in[3];
  const float* bk = (const float*)d_in[4];
  const float* Wv = (const float*)d_in[5];
  const float* bv = (const float*)d_in[6];
  const float* Wp = (const float*)d_in[7];
  const float* bp = (const float*)d_in[8];
  float* out = (float*)d_out;

  // Workspace layout (floats): attn | pooled | q | k | A | beta  (~277 MB)
  float* ws     = (float*)d_ws;
  float* attn   = ws;                                    // B*C*H*W = 67,108,864
  float* pooled = attn   + (size_t)BN * CN * HWN;        // 65,536
  float* qb     = pooled + (size_t)BN * CN * 64;         // 524,288
  float* kb     = qb     + (size_t)BN * CN * 512;        // 524,288
  float* Ab     = kb     + (size_t)BN * CN * 512;        // 1,048,576
  float* betab  = Ab     + (size_t)BN * HEADSN * 128 * 128;  // 8,192

  k_pool<<<dim3(BN * CN), dim3(256), 0, stream>>>(x, pooled);
  k_qk<<<dim3((BN * CN * 512) / 256), dim3(256), 0, stream>>>(pooled, Wq, bq, Wk, bk, qb, kb);
  k_prep<<<dim3(BN * HEADSN), dim3(256), 0, stream>>>(qb, kb, Wv, bv, Ab, betab);
  k_attn<<<dim3(D2N / 64, BN * HEADSN), dim3(256), 0, stream>>>(x, Ab, betab, attn);
  k_dw<<<dim3((size_t)(BN * CN * HWN) / 4 / 256), dim3(256), 0, stream>>>(attn, Wp, bp, out);
}